// channel_spatial_attention_42090679501608
// MI455X (gfx1250) — compile-verified
//
#include <hip/hip_runtime.h>

typedef __attribute__((ext_vector_type(2))) float v2f;
typedef __attribute__((ext_vector_type(8))) float v8f;

#define CDIM 128
#define HDIM 128
#define WDIM 128
#define BDIM 8
#define HW (HDIM * WDIM)

// ---------------- kernel 1: global average pool per (b,c) ----------------
__global__ __launch_bounds__(256) void pool_kernel(const float* __restrict__ x,
                                                   float* __restrict__ pool) {
  const int bc = blockIdx.x;                 // 0..1023
  const float* p = x + (size_t)bc * HW;
  float s = 0.f;
  for (int i = threadIdx.x; i < HW; i += 256) s += p[i];
  __shared__ float red[256];
  red[threadIdx.x] = s;
  __syncthreads();
  for (int off = 128; off > 0; off >>= 1) {
    if ((int)threadIdx.x < off) red[threadIdx.x] += red[threadIdx.x + off];
    __syncthreads();
  }
  if (threadIdx.x == 0) pool[bc] = red[0] * (1.0f / (float)HW);
}

// ------ kernel 2: addcs[b,c] = b_sp[c] + b_ch + conv1d(pool, w_ch)[b,c] ------
__global__ __launch_bounds__(256) void addcs_kernel(const float* __restrict__ pool,
                                                    const float* __restrict__ b_sp,
                                                    const float* __restrict__ w_ch,
                                                    const float* __restrict__ b_ch,
                                                    float* __restrict__ addcs) {
  const int idx = blockIdx.x * 256 + threadIdx.x;
  if (idx >= BDIM * CDIM) return;
  const int b = idx >> 7, c = idx & (CDIM - 1);
  float s = b_ch[0];
#pragma unroll
  for (int d = 0; d < 3; ++d) {
    const int cc = c + d - 1;
    if (cc >= 0 && cc < CDIM) s += w_ch[d] * pool[b * CDIM + cc];
  }
  addcs[idx] = s + b_sp[c];
}

// ---------------- main fused kernel ----------------
// Grid: B * H * (W/16) blocks, 256 threads (8 waves of 32).
// Per block: 16 pixels of one image row.
//   Stage 1: 128ch x 3row x 18col halo x-tile -> LDS
//   Stage 2: cs[p][c] = depthwise3x3 + addcs            (A matrix, M=16, K=128)
//   Stage 3: per wave: 9 WMMA N-tiles (o-range 144 = 16 channels x 9 taps),
//            32 chained V_WMMA_F32_16X16X4_F32 per tile (K=128), +bias -> LDS
//   Stage 4: out[b,c,h,w] = sum_k f[p][c*9+k] * xtile[c][tap k]
__global__ __launch_bounds__(256) void fused_main(const float* __restrict__ x,
                                                  const float* __restrict__ w_sp,
                                                  const float* __restrict__ w_conv,
                                                  const float* __restrict__ b_conv,
                                                  const float* __restrict__ addcs,
                                                  float* __restrict__ out) {
  extern __shared__ char smem_raw[];
  float(*xt)[3][18] = reinterpret_cast<float(*)[3][18]>(smem_raw);                  // [128][3][18] 27648B
  float(*csL)[132] = reinterpret_cast<float(*)[132]>(smem_raw + 27648);             // [16][132]     8448B
  float(*fbuf)[16][144] = reinterpret_cast<float(*)[16][144]>(smem_raw + 36096);    // [8][16][144] 73728B

  const int blk = blockIdx.x;            // 8 * 128 * 8 = 8192
  const int b = blk >> 10;
  const int h = (blk >> 3) & 127;
  const int w0 = (blk & 7) << 4;
  const int tid = threadIdx.x;
  const int lane = tid & 31;
  const int wv = tid >> 5;

  const size_t base_b = (size_t)b * CDIM * HW;

  // ---- Stage 1: load halo tile (zero padded at image borders) ----
  for (int e = tid; e < CDIM * 3 * 18; e += 256) {
    const int c = e / 54;
    const int rem = e - c * 54;
    const int r = rem / 18;
    const int col = rem - r * 18;
    const int hh = h + r - 1;
    const int ww = w0 + col - 1;
    float v = 0.f;
    if ((unsigned)hh < (unsigned)HDIM && (unsigned)ww < (unsigned)WDIM)
      v = x[base_b + ((size_t)c * HDIM + hh) * WDIM + ww];
    xt[c][r][col] = v;
  }
  __syncthreads();

  // ---- Stage 2: cs (depthwise 3x3 + per-(b,c) additive term) ----
  {
    const int c = tid >> 1;              // 2 threads per channel
    const int p0 = (tid & 1) << 3;       // 8 pixels each
    float w9[9];
#pragma unroll
    for (int i = 0; i < 9; ++i) w9[i] = w_sp[c * 9 + i];
    const float add = addcs[b * CDIM + c];
#pragma unroll
    for (int pp = 0; pp < 8; ++pp) {
      const int p = p0 + pp;
      float s = add;
#pragma unroll
      for (int r = 0; r < 3; ++r)
#pragma unroll
        for (int j = 0; j < 3; ++j) s += w9[r * 3 + j] * xt[c][r][p + j];
      csL[p][c] = s;
    }
  }
  __syncthreads();

  // ---- Stage 3: per-wave GEMM over its 144-wide o-group ----
  // WMMA f32 16x16x4 lane mapping (ISA 7.12.2):
  //   A: row m=lane&15, K-pair kb=2*(lane>>4); B: col n=lane&15, same K-pair
  //   D: element v -> D[v + 8*(lane>>4)][lane&15]
  const int m = lane & 15;
  const int n = lane & 15;
  const int kb = (lane >> 4) << 1;
  const int hi = (lane >> 4) << 3;

  v2f afrag[32];
#pragma unroll
  for (int t = 0; t < 32; ++t)
    afrag[t] = *reinterpret_cast<const v2f*>(&csL[m][4 * t + kb]);

#pragma unroll 1
  for (int nt = 0; nt < 9; ++nt) {
    const int nb = wv * 144 + nt * 16;
    const float* Bp = w_conv + (size_t)(nb + n) * CDIM + kb;
    if (nt < 8) __builtin_prefetch(w_conv + (size_t)(nb + 16 + n) * CDIM, 0, 1);
    v8f d = {0.f, 0.f, 0.f, 0.f, 0.f, 0.f, 0.f, 0.f};
#pragma unroll
    for (int t = 0; t < 32; ++t) {
      const v2f bfrag = *reinterpret_cast<const v2f*>(Bp + 4 * t);
      d = __builtin_amdgcn_wmma_f32_16x16x4_f32(false, afrag[t], false, bfrag,
                                                (short)0, d, false, false);
    }
    const float bias = b_conv[nb + n];
#pragma unroll
    for (int v = 0; v < 8; ++v)
      fbuf[wv][v + hi][nt * 16 + n] = d[v] + bias;
  }
  __syncthreads();

  // ---- Stage 4: modulate with 3x3 patches, reduce 9 taps, store ----
  {
    const int p = lane & 15;
    const size_t obase = base_b + (size_t)h * WDIM + (w0 + p);
#pragma unroll
    for (int i = 0; i < 8; ++i) {
      const int cl = hi + i;             // 0..15 within group
      const int c = wv * 16 + cl;
      float acc = 0.f;
#pragma unroll
      for (int k = 0; k < 9; ++k)
        acc += fbuf[wv][p][cl * 9 + k] * xt[c][k / 3][p + (k % 3)];
      out[obase + (size_t)c * HW] = acc;
    }
  }
}

extern "C" void kernel_launch(void* const* d_in, const int* in_sizes, int n_in,
                              void* d_out, int out_size, void* d_ws, size_t ws_size,
                              hipStream_t stream) {
  const float* x = (const float*)d_in[0];
  const float* w_sp = (const float*)d_in[1];
  const float* b_sp = (const float*)d_in[2];
  const float* w_ch = (const float*)d_in[3];
  const float* b_ch = (const float*)d_in[4];
  const float* w_conv = (const float*)d_in[5];
  const float* b_conv = (const float*)d_in[6];
  float* out = (float*)d_out;

  float* pool = (float*)d_ws;                 // 1024 floats
  float* addcs = pool + BDIM * CDIM;          // 1024 floats

  pool_kernel<<<BDIM * CDIM, 256, 0, stream>>>(x, pool);
  addcs_kernel<<<(BDIM * CDIM + 255) / 256, 256, 0, stream>>>(pool, b_sp, w_ch,
                                                              b_ch, addcs);

  const size_t lds_bytes = 27648 + 8448 + 73728;  // 109824
  fused_main<<<BDIM * HDIM * (WDIM / 16), 256, lds_bytes, stream>>>(
      x, w_sp, w_conv, b_conv, addcs, out);
}